// SLayerExponential_77747497992595
// MI455X (gfx1250) — compile-verified
//
#include <hip/hip_runtime.h>

// SLayer exponential: out[b,n] = sum_p nd[b,p] * exp(-sum_d s2[n,d]*(c[n,d]-x[b,p,d])^2)
// K=8 GEMM via V_WMMA_F32_16X16X4_F32 x2. log2(e) folded into cluster weights and
// log2(not_dummy) folded into the K=7 feature row, so the epilogue per element is
// exactly one v_exp_f32 (2^x) + one add. No LDS traffic at all.

typedef float v2f __attribute__((ext_vector_type(2)));
typedef float v8f __attribute__((ext_vector_type(8)));

#define B_  64
#define P_  2048
#define N_  512
#define NT_ (N_ / 16)   // 32 n-tiles of 16 clusters

template <int SPLIT>
__global__ __launch_bounds__(256) void slayer_wmma_kernel(
    const float* __restrict__ x,          // [B, P, 3]
    const float* __restrict__ not_dummy,  // [B, P]
    const float* __restrict__ centers,    // [N, 3]
    const float* __restrict__ sharpness,  // [N, 3]
    float* __restrict__ part)             // [SPLIT, B, N]  (== out when SPLIT==1)
{
    const int lane = threadIdx.x & 31;
    const int wid  = blockIdx.x * (blockDim.x >> 5) + (threadIdx.x >> 5);
    // wid = ((b * NT_ + nt) * SPLIT + s)
    const int s    = wid % SPLIT;
    const int tid  = wid / SPLIT;
    const int nt   = tid % NT_;
    const int b    = tid / NT_;

    const int  ln = lane & 15;
    const bool hi = lane >= 16;

    // ---- B fragment: per-cluster weights, pre-scaled by log2(e) so the WMMA
    // output is log2( nd * exp(-dist) ) once the K=7 row adds log2(nd). ----
    const int   n  = nt * 16 + ln;
    const float c0 = centers[n * 3 + 0], c1 = centers[n * 3 + 1], c2 = centers[n * 3 + 2];
    const float h0 = sharpness[n * 3 + 0], h1 = sharpness[n * 3 + 1], h2 = sharpness[n * 3 + 2];
    const float LOG2E = 1.4426950408889634f;
    const float s0 = h0 * h0 * LOG2E, s1 = h1 * h1 * LOG2E, s2 = h2 * h2 * LOG2E;
    const float w0 = 2.0f * s0 * c0, w1 = 2.0f * s1 * c1, w2 = 2.0f * s2 * c2; // vs x_d
    const float w3 = -s0, w4 = -s1, w5 = -s2;                                   // vs x_d^2
    const float w6 = -(s0 * c0 * c0 + s1 * c1 * c1 + s2 * c2 * c2);             // vs 1
    // ISA 32-bit B 4x16 layout: VGPR0 = {K0 | K2}, VGPR1 = {K1 | K3} across half-waves.
    v2f bf0, bf1;
    bf0.x = hi ? w2 : w0;
    bf0.y = hi ? w3 : w1;
    bf1.x = hi ? w6 : w4;    // K=6 row: constant term
    bf1.y = hi ? 1.0f : w5;  // K=7 row: weight 1.0 multiplies log2(nd[p])

    const int    prows = P_ / SPLIT;
    const int    p0    = s * prows;
    const float* xb    = x + (size_t)b * P_ * 3;
    const float* ndb   = not_dummy + (size_t)b * P_;

    v8f acc = {};   // 8 independent accumulators -> no serial add chain

    for (int pt = 0; pt < prows; pt += 16) {
        const int    p  = p0 + pt;
        const float* xr = xb + (size_t)(p + ln) * 3;
        const float  x0 = xr[0], x1 = xr[1], x2 = xr[2];
        const float  lg = __builtin_amdgcn_logf(ndb[p + ln]);  // log2(nd); 0 -> -inf

        // ISA 32-bit A 16x4 layout: VGPR0 = {K0 | K2}, VGPR1 = {K1 | K3}.
        v2f a0, a1;
        a0.x = hi ? x2 : x0;          // f2 | f0
        a0.y = hi ? x0 * x0 : x1;     // f3 | f1
        a1.x = hi ? 1.0f : x1 * x1;   // f6 | f4
        a1.y = hi ? lg : x2 * x2;     // f7 | f5

        v8f c = {};
        c = __builtin_amdgcn_wmma_f32_16x16x4_f32(false, a0, false, bf0,
                                                  (short)0, c, false, false);
        c = __builtin_amdgcn_wmma_f32_16x16x4_f32(false, a1, false, bf1,
                                                  (short)0, c, false, false);

#pragma unroll
        for (int r = 0; r < 8; ++r) {
            acc[r] += __builtin_amdgcn_exp2f(c[r]);  // one v_exp_f32 + one add
        }
    }

    float total = ((acc[0] + acc[1]) + (acc[2] + acc[3])) +
                  ((acc[4] + acc[5]) + (acc[6] + acc[7]));

    // Lane l and l+16 hold rows 0-7 / 8-15 of the same column n: combine.
    total += __shfl_xor(total, 16, 32);
    if (!hi) {
        part[(size_t)s * (B_ * N_) + (size_t)b * N_ + n] = total;
    }
}

__global__ __launch_bounds__(256) void slayer_reduce_kernel(
    const float* __restrict__ part, float* __restrict__ out, int count, int split)
{
    const int i = blockIdx.x * blockDim.x + threadIdx.x;
    if (i < count) {
        float s = 0.0f;
        for (int k = 0; k < split; ++k) s += part[(size_t)k * count + i];
        out[i] = s;
    }
}

extern "C" void kernel_launch(void* const* d_in, const int* in_sizes, int n_in,
                              void* d_out, int out_size, void* d_ws, size_t ws_size,
                              hipStream_t stream) {
    const float* x       = (const float*)d_in[0];  // [64, 2048, 3]
    const float* nd      = (const float*)d_in[1];  // [64, 2048]
    const float* centers = (const float*)d_in[2];  // [512, 3]
    const float* sharp   = (const float*)d_in[3];  // [512, 3]
    float*       out     = (float*)d_out;          // [64, 512]

    constexpr int SPLIT = 4;
    const size_t need = (size_t)SPLIT * B_ * N_ * sizeof(float);

    if (d_ws != nullptr && ws_size >= need) {
        float* part = (float*)d_ws;
        const int waves  = B_ * NT_ * SPLIT;  // 8192 waves
        const int blocks = waves / 8;         // 256 threads = 8 wave32 per block
        slayer_wmma_kernel<SPLIT><<<blocks, 256, 0, stream>>>(x, nd, centers, sharp, part);
        const int count = B_ * N_;
        slayer_reduce_kernel<<<(count + 255) / 256, 256, 0, stream>>>(part, out, count, SPLIT);
    } else {
        const int waves  = B_ * NT_;          // 2048 waves, direct write
        const int blocks = waves / 8;
        slayer_wmma_kernel<1><<<blocks, 256, 0, stream>>>(x, nd, centers, sharp, out);
    }
}